// MoNet_78323023610196
// MI455X (gfx1250) — compile-verified
//
#include <hip/hip_runtime.h>
#include <math.h>

// ---- problem constants (fixed by the reference) ----
#define N0   30720
#define E0   184320
#define N1C  21504
#define E1C  129024
#define N2C  15360
#define E2C  92160
#define NG   256
#define KK   25

typedef float v2f __attribute__((ext_vector_type(2)));
typedef float v8f __attribute__((ext_vector_type(8)));

// ---------------- workspace layout (float offsets) ----------------
// zero-initialized region first (one k_zero launch covers it)
static constexpr size_t F_AGG1 = 0;
static constexpr size_t F_CNT1 = F_AGG1 + (size_t)N0 * 32;
static constexpr size_t F_H1   = F_CNT1 + (size_t)N0;
static constexpr size_t F_P1S  = F_H1   + (size_t)N1C * 32;
static constexpr size_t F_C1A  = F_P1S  + (size_t)N1C * 2;
static constexpr size_t F_MAX1 = F_C1A  + (size_t)N1C;
static constexpr size_t F_AGG2 = F_MAX1 + 64;
static constexpr size_t F_CNT2 = F_AGG2 + (size_t)N1C * 64;
static constexpr size_t F_H2   = F_CNT2 + (size_t)N1C;
static constexpr size_t F_P2S  = F_H2   + (size_t)N2C * 64;
static constexpr size_t F_C2A  = F_P2S  + (size_t)N2C * 2;
static constexpr size_t F_MAX2 = F_C2A  + (size_t)N2C;
static constexpr size_t F_AGG3 = F_MAX2 + 64;
static constexpr size_t F_CNT3 = F_AGG3 + (size_t)N2C * 64;
static constexpr size_t F_ZSUM = F_CNT3 + (size_t)N2C;
static constexpr size_t F_ZCNT = F_ZSUM + (size_t)NG * 64;
static constexpr size_t F_ZERO_END = F_ZCNT + (size_t)NG;
// non-zeroed scratch (fully overwritten before reads)
static constexpr size_t F_GAUSS = F_ZERO_END;                    // reused L1/L2/L3
static constexpr size_t F_HA    = F_GAUSS + (size_t)E0 * KK;
static constexpr size_t F_POS1  = F_HA    + (size_t)N0 * 32;
static constexpr size_t F_CART1 = F_POS1  + (size_t)N1C * 2;
static constexpr size_t F_XG    = F_CART1 + (size_t)E1C * 2;     // reused xg2 / xg3
static constexpr size_t F_RT2   = F_XG    + (size_t)N1C * 1600;
static constexpr size_t F_HB    = F_RT2   + (size_t)N1C * 64;
static constexpr size_t F_POS2  = F_HB    + (size_t)N1C * 64;
static constexpr size_t F_CART2 = F_POS2  + (size_t)N2C * 2;
static constexpr size_t F_RT3   = F_CART2 + (size_t)E2C * 2;
static constexpr size_t F_HC    = F_RT3   + (size_t)N2C * 64;
static constexpr size_t F_ZMEAN = F_HC    + (size_t)N2C * 64;
static constexpr size_t F_Z1    = F_ZMEAN + (size_t)NG * 64;

// ---------------- kernels ----------------
__global__ void k_zero(float* __restrict__ p, int n) {
    int i = blockIdx.x * blockDim.x + threadIdx.x;
    if (i < n) p[i] = 0.0f;
}

// Gaussian kernel weights from precomputed edge attrs (layer 1)
__global__ void k_gauss_ea(const float* __restrict__ ea, const float* __restrict__ mu,
                           const float* __restrict__ sg, float* __restrict__ gauss, int E) {
    int e = blockIdx.x * blockDim.x + threadIdx.x;
    if (e >= E) return;
    float ax = ea[e * 2], ay = ea[e * 2 + 1];
#pragma unroll
    for (int k = 0; k < KK; k++) {
        float dx = ax - mu[k * 2], dy = ay - mu[k * 2 + 1];
        float sx = sg[k * 2], sy = sg[k * 2 + 1];
        float t = -0.5f * (dx * dx / (1e-15f + sx * sx) + dy * dy / (1e-15f + sy * sy));
        gauss[(size_t)e * KK + k] = expf(t);
    }
}

// per-edge cart = pos[s]-pos[d]; block-reduced global max |cart|
__global__ void k_cartmax(const float* __restrict__ pos, const int* __restrict__ ei,
                          float* __restrict__ cart, unsigned int* __restrict__ gmax, int E) {
    __shared__ float sm[256];
    int e = blockIdx.x * blockDim.x + threadIdx.x;
    float loc = 0.0f;
    if (e < E) {
        int s = ei[e], d = ei[E + e];
        float cx = pos[s * 2] - pos[d * 2];
        float cy = pos[s * 2 + 1] - pos[d * 2 + 1];
        cart[e * 2] = cx; cart[e * 2 + 1] = cy;
        loc = fmaxf(fabsf(cx), fabsf(cy));
    }
    sm[threadIdx.x] = loc;
    __syncthreads();
    for (int st = 128; st > 0; st >>= 1) {
        if (threadIdx.x < st) sm[threadIdx.x] = fmaxf(sm[threadIdx.x], sm[threadIdx.x + st]);
        __syncthreads();
    }
    if (threadIdx.x == 0) atomicMax(gmax, __float_as_uint(sm[0]));
}

// gauss from stored cart, normalized by global max (layers 2/3)
__global__ void k_gauss_cart(const float* __restrict__ cart, const unsigned int* __restrict__ gmax,
                             const float* __restrict__ mu, const float* __restrict__ sg,
                             float* __restrict__ gauss, int E) {
    int e = blockIdx.x * blockDim.x + threadIdx.x;
    if (e >= E) return;
    float sc = 0.5f / fmaxf(__uint_as_float(*gmax), 1e-12f);
    float ax = cart[e * 2] * sc + 0.5f, ay = cart[e * 2 + 1] * sc + 0.5f;
#pragma unroll
    for (int k = 0; k < KK; k++) {
        float dx = ax - mu[k * 2], dy = ay - mu[k * 2 + 1];
        float sx = sg[k * 2], sy = sg[k * 2 + 1];
        float t = -0.5f * (dx * dx / (1e-15f + sx * sx) + dy * dy / (1e-15f + sy * sy));
        gauss[(size_t)e * KK + k] = expf(t);
    }
}

// layer 1 fused edge pass: K_in=3, O=32, thread = (edge, o)
__global__ void k_edge1(const int* __restrict__ ei, const float* __restrict__ x,
                        const float* __restrict__ pos, const float* __restrict__ g1,
                        const float* __restrict__ gauss, float* __restrict__ agg,
                        float* __restrict__ cnt) {
    int t = blockIdx.x * blockDim.x + threadIdx.x;
    if (t >= E0 * 32) return;
    int e = t >> 5, o = t & 31;
    int s = ei[e], d = ei[E0 + e];
    float a0 = pos[s * 2], a1 = pos[s * 2 + 1], a2 = x[s];
    const float* gp = gauss + (size_t)e * KK;
    float msg = 0.0f;
#pragma unroll
    for (int k = 0; k < KK; k++) {
        float xgv = a0 * g1[k * 32 + o] + a1 * g1[800 + k * 32 + o] + a2 * g1[1600 + k * 32 + o];
        msg += gp[k] * xgv;
    }
    atomicAdd(&agg[(size_t)d * 32 + o], msg);
    if (o == 0) atomicAdd(&cnt[d], 1.0f);
}

__global__ void k_node1(const float* __restrict__ x, const float* __restrict__ pos,
                        const float* __restrict__ agg, const float* __restrict__ cnt,
                        const float* __restrict__ root, const float* __restrict__ b,
                        float* __restrict__ out) {
    int t = blockIdx.x * blockDim.x + threadIdx.x;
    if (t >= N0 * 32) return;
    int n = t >> 5, o = t & 31;
    float v = agg[t] / fmaxf(cnt[n], 1.0f)
            + pos[n * 2] * root[o] + pos[n * 2 + 1] * root[32 + o] + x[n] * root[64 + o] + b[o];
    out[t] = fmaxf(v, 0.0f);
}

// segment max pooling (src >= 0, dst zero-initialized) via uint-bit atomicMax
__global__ void k_poolmax(const float* __restrict__ src, const int* __restrict__ cluster,
                          float* __restrict__ dst, int N, int O) {
    int t = blockIdx.x * blockDim.x + threadIdx.x;
    if (t >= N * O) return;
    int n = t / O, o = t - n * O;
    atomicMax((unsigned int*)&dst[(size_t)cluster[n] * O + o], __float_as_uint(src[t]));
}

__global__ void k_posacc(const float* __restrict__ pos, const int* __restrict__ cluster,
                         float* __restrict__ psum, float* __restrict__ pcnt, int N) {
    int n = blockIdx.x * blockDim.x + threadIdx.x;
    if (n >= N) return;
    int c = cluster[n];
    atomicAdd(&psum[c * 2], pos[n * 2]);
    atomicAdd(&psum[c * 2 + 1], pos[n * 2 + 1]);
    atomicAdd(&pcnt[c], 1.0f);
}

__global__ void k_posdiv(const float* __restrict__ psum, const float* __restrict__ pcnt,
                         float* __restrict__ pout, int N) {
    int n = blockIdx.x * blockDim.x + threadIdx.x;
    if (n >= N) return;
    float inv = 1.0f / fmaxf(pcnt[n], 1.0f);
    pout[n * 2] = psum[n * 2] * inv;
    pout[n * 2 + 1] = psum[n * 2 + 1] * inv;
}

// fp32 WMMA GEMM: C(MxN) = A(MxK) @ B(KxN); M,N % 16 == 0, K % 4 == 0.
// One wave32 per 16x16 C tile using V_WMMA_F32_16X16X4_F32.
// Operand layouts per CDNA5 ISA 7.12.2 (lane halves carry K/M offsets).
__global__ void k_wmma_gemm(const float* __restrict__ A, const float* __restrict__ B,
                            float* __restrict__ C, int M, int N, int K) {
    int wave = blockIdx.x * (blockDim.x >> 5) + (threadIdx.x >> 5);
    int tilesN = N >> 4;
    int nTiles = (M >> 4) * tilesN;
    if (wave >= nTiles) return;               // whole-wave uniform exit: EXEC stays all-1
    int tm = wave / tilesN, tn = wave - tm * tilesN;
    int m0 = tm << 4, n0 = tn << 4;
    int lane = threadIdx.x & 31;
    int half = lane >> 4, l = lane & 15;
    v8f acc = {0.f, 0.f, 0.f, 0.f, 0.f, 0.f, 0.f, 0.f};
    const float* Arow = A + (size_t)(m0 + l) * K;
    for (int k = 0; k < K; k += 4) {
        int ka = k + (half << 1);
        v2f a, b;
        a.x = Arow[ka];
        a.y = Arow[ka + 1];
        b.x = B[(size_t)ka * N + n0 + l];
        b.y = B[(size_t)(ka + 1) * N + n0 + l];
        acc = __builtin_amdgcn_wmma_f32_16x16x4_f32(false, a, false, b, (short)0, acc, false, false);
    }
    float* Crow = C + (size_t)(m0 + (half << 3)) * N + n0 + l;
#pragma unroll
    for (int r = 0; r < 8; r++) Crow[(size_t)r * N] = acc[r];
}

// layers 2/3 edge pass: O=64, xg row stride 1600 (25*64), thread = (edge, o)
__global__ void k_edge2(const int* __restrict__ ei, const unsigned char* __restrict__ em,
                        const float* __restrict__ gauss, const float* __restrict__ xg,
                        float* __restrict__ agg, float* __restrict__ cnt, int E) {
    int t = blockIdx.x * blockDim.x + threadIdx.x;
    if (t >= E * 64) return;
    int e = t >> 6, o = t & 63;
    if (!em[e]) return;                       // w=0 edges contribute nothing
    int s = ei[e], d = ei[E + e];
    const float* xgp = xg + (size_t)s * 1600 + o;
    const float* gp = gauss + (size_t)e * KK;
    float msg = 0.0f;
#pragma unroll
    for (int k = 0; k < KK; k++) msg += gp[k] * xgp[k * 64];
    atomicAdd(&agg[(size_t)d * 64 + o], msg);
    if (o == 0) atomicAdd(&cnt[d], 1.0f);
}

__global__ void k_nodeupd(const float* __restrict__ agg, const float* __restrict__ cnt,
                          const float* __restrict__ rootT, const float* __restrict__ b,
                          float* __restrict__ out, int N, int relu) {
    int t = blockIdx.x * blockDim.x + threadIdx.x;
    if (t >= N * 64) return;
    int n = t >> 6, o = t & 63;
    float v = agg[t] / fmaxf(cnt[n], 1.0f) + rootT[t] + b[o];
    out[t] = relu ? fmaxf(v, 0.0f) : v;
}

__global__ void k_zacc(const float* __restrict__ h, const unsigned char* __restrict__ nm,
                       const int* __restrict__ batch, float* __restrict__ zsum,
                       float* __restrict__ zcnt) {
    int t = blockIdx.x * blockDim.x + threadIdx.x;
    if (t >= N2C * 64) return;
    int n = t >> 6, o = t & 63;
    if (!nm[n]) return;
    int g = batch[n];
    atomicAdd(&zsum[(size_t)g * 64 + o], h[t]);
    if (o == 0) atomicAdd(&zcnt[g], 1.0f);
}

__global__ void k_zdiv(const float* __restrict__ zsum, const float* __restrict__ zcnt,
                       float* __restrict__ z) {
    int t = blockIdx.x * blockDim.x + threadIdx.x;
    if (t >= NG * 64) return;
    z[t] = zsum[t] / fmaxf(zcnt[t >> 6], 1.0f);
}

__global__ void k_biasrelu(float* __restrict__ z, const float* __restrict__ b) {
    int t = blockIdx.x * blockDim.x + threadIdx.x;
    if (t >= NG * 128) return;
    z[t] = fmaxf(z[t] + b[t & 127], 0.0f);
}

__global__ void k_final(const float* __restrict__ z, const float* __restrict__ w,
                        const float* __restrict__ b, float* __restrict__ out) {
    int g = blockIdx.x * blockDim.x + threadIdx.x;
    if (g >= NG) return;
    float l[10];
#pragma unroll
    for (int c = 0; c < 10; c++) {
        float a = b[c];
        for (int j = 0; j < 128; j++) a += z[g * 128 + j] * w[j * 10 + c];
        l[c] = a;
    }
    float m = l[0];
#pragma unroll
    for (int c = 1; c < 10; c++) m = fmaxf(m, l[c]);
    float s = 0.0f;
#pragma unroll
    for (int c = 0; c < 10; c++) s += expf(l[c] - m);
    float ls = logf(s);
#pragma unroll
    for (int c = 0; c < 10; c++) out[g * 10 + c] = l[c] - m - ls;
}

// ---------------- launch ----------------
static inline int grd(long long n) { return (int)((n + 255) / 256); }

extern "C" void kernel_launch(void* const* d_in, const int* in_sizes, int n_in,
                              void* d_out, int out_size, void* d_ws, size_t ws_size,
                              hipStream_t stream) {
    (void)in_sizes; (void)n_in; (void)out_size; (void)ws_size;
    const float* x    = (const float*)d_in[0];
    const float* pos  = (const float*)d_in[1];
    const float* ea0  = (const float*)d_in[2];
    const int*   ei0  = (const int*)d_in[3];
    const int*   c1   = (const int*)d_in[4];
    const int*   ei1  = (const int*)d_in[5];
    const unsigned char* em1 = (const unsigned char*)d_in[6];
    const int*   c2   = (const int*)d_in[8];
    const int*   ei2  = (const int*)d_in[9];
    const unsigned char* em2 = (const unsigned char*)d_in[10];
    const unsigned char* nm2 = (const unsigned char*)d_in[11];
    const int*   bt2  = (const int*)d_in[12];
    const float* g1   = (const float*)d_in[14];
    const float* mu1  = (const float*)d_in[15];
    const float* sg1  = (const float*)d_in[16];
    const float* r1   = (const float*)d_in[17];
    const float* b1   = (const float*)d_in[18];
    const float* g2   = (const float*)d_in[19];
    const float* mu2  = (const float*)d_in[20];
    const float* sg2  = (const float*)d_in[21];
    const float* r2   = (const float*)d_in[22];
    const float* b2   = (const float*)d_in[23];
    const float* g3   = (const float*)d_in[24];
    const float* mu3  = (const float*)d_in[25];
    const float* sg3  = (const float*)d_in[26];
    const float* r3   = (const float*)d_in[27];
    const float* b3   = (const float*)d_in[28];
    const float* wl1  = (const float*)d_in[29];
    const float* bl1  = (const float*)d_in[30];
    const float* wl2  = (const float*)d_in[31];
    const float* bl2  = (const float*)d_in[32];
    float* W = (float*)d_ws;
    float* out = (float*)d_out;

    // zero all accumulators in one shot
    k_zero<<<grd(F_ZERO_END), 256, 0, stream>>>(W, (int)F_ZERO_END);

    // ---- layer 1 conv (on [pos, x], O=32) ----
    k_gauss_ea<<<grd(E0), 256, 0, stream>>>(ea0, mu1, sg1, W + F_GAUSS, E0);
    k_edge1<<<grd((long long)E0 * 32), 256, 0, stream>>>(ei0, x, pos, g1, W + F_GAUSS,
                                                         W + F_AGG1, W + F_CNT1);
    k_node1<<<grd((long long)N0 * 32), 256, 0, stream>>>(x, pos, W + F_AGG1, W + F_CNT1,
                                                         r1, b1, W + F_HA);
    // ---- pool 1 ----
    k_poolmax<<<grd((long long)N0 * 32), 256, 0, stream>>>(W + F_HA, c1, W + F_H1, N0, 32);
    k_posacc<<<grd(N0), 256, 0, stream>>>(pos, c1, W + F_P1S, W + F_C1A, N0);
    k_posdiv<<<grd(N1C), 256, 0, stream>>>(W + F_P1S, W + F_C1A, W + F_POS1, N1C);

    // ---- layer 2 conv (O=64) ----
    k_cartmax<<<grd(E1C), 256, 0, stream>>>(W + F_POS1, ei1, W + F_CART1,
                                            (unsigned int*)(W + F_MAX1), E1C);
    k_gauss_cart<<<grd(E1C), 256, 0, stream>>>(W + F_CART1, (unsigned int*)(W + F_MAX1),
                                               mu2, sg2, W + F_GAUSS, E1C);
    {   // xg2 = h1 @ g2  (21504x32 @ 32x1600), rootT2 = h1 @ root2
        int t1 = (N1C / 16) * (1600 / 16);
        k_wmma_gemm<<<(t1 + 7) / 8, 256, 0, stream>>>(W + F_H1, g2, W + F_XG, N1C, 1600, 32);
        int t2 = (N1C / 16) * (64 / 16);
        k_wmma_gemm<<<(t2 + 7) / 8, 256, 0, stream>>>(W + F_H1, r2, W + F_RT2, N1C, 64, 32);
    }
    k_edge2<<<grd((long long)E1C * 64), 256, 0, stream>>>(ei1, em1, W + F_GAUSS, W + F_XG,
                                                          W + F_AGG2, W + F_CNT2, E1C);
    k_nodeupd<<<grd((long long)N1C * 64), 256, 0, stream>>>(W + F_AGG2, W + F_CNT2, W + F_RT2,
                                                            b2, W + F_HB, N1C, 1);
    // ---- pool 2 ----
    k_poolmax<<<grd((long long)N1C * 64), 256, 0, stream>>>(W + F_HB, c2, W + F_H2, N1C, 64);
    k_posacc<<<grd(N1C), 256, 0, stream>>>(W + F_POS1, c2, W + F_P2S, W + F_C2A, N1C);
    k_posdiv<<<grd(N2C), 256, 0, stream>>>(W + F_P2S, W + F_C2A, W + F_POS2, N2C);

    // ---- layer 3 conv (no relu) ----
    k_cartmax<<<grd(E2C), 256, 0, stream>>>(W + F_POS2, ei2, W + F_CART2,
                                            (unsigned int*)(W + F_MAX2), E2C);
    k_gauss_cart<<<grd(E2C), 256, 0, stream>>>(W + F_CART2, (unsigned int*)(W + F_MAX2),
                                               mu3, sg3, W + F_GAUSS, E2C);
    {   // xg3 = h2 @ g3  (15360x64 @ 64x1600), rootT3 = h2 @ root3
        int t1 = (N2C / 16) * (1600 / 16);
        k_wmma_gemm<<<(t1 + 7) / 8, 256, 0, stream>>>(W + F_H2, g3, W + F_XG, N2C, 1600, 64);
        int t2 = (N2C / 16) * (64 / 16);
        k_wmma_gemm<<<(t2 + 7) / 8, 256, 0, stream>>>(W + F_H2, r3, W + F_RT3, N2C, 64, 64);
    }
    k_edge2<<<grd((long long)E2C * 64), 256, 0, stream>>>(ei2, em2, W + F_GAUSS, W + F_XG,
                                                          W + F_AGG3, W + F_CNT3, E2C);
    k_nodeupd<<<grd((long long)N2C * 64), 256, 0, stream>>>(W + F_AGG3, W + F_CNT3, W + F_RT3,
                                                            b3, W + F_HC, N2C, 0);

    // ---- global masked mean + MLP head ----
    k_zacc<<<grd((long long)N2C * 64), 256, 0, stream>>>(W + F_HC, nm2, bt2,
                                                         W + F_ZSUM, W + F_ZCNT);
    k_zdiv<<<grd(NG * 64), 256, 0, stream>>>(W + F_ZSUM, W + F_ZCNT, W + F_ZMEAN);
    {   // z1 = zmean @ w_lin1 (256x64 @ 64x128)
        int t1 = (NG / 16) * (128 / 16);
        k_wmma_gemm<<<(t1 + 7) / 8, 256, 0, stream>>>(W + F_ZMEAN, wl1, W + F_Z1, NG, 128, 64);
    }
    k_biasrelu<<<grd(NG * 128), 256, 0, stream>>>(W + F_Z1, bl1);
    k_final<<<grd(NG), 256, 0, stream>>>(W + F_Z1, wl2, bl2, out);
}